// GCNModel_63488206570136
// MI455X (gfx1250) — compile-verified
//
#include <hip/hip_runtime.h>
#include <hip/hip_bf16.h>

typedef float v2f __attribute__((ext_vector_type(2)));
typedef float v8f __attribute__((ext_vector_type(8)));

#define F_IN 128
#define HID  64

// ---------------- degree / norm prep ----------------

__global__ void k_init_deg(float* deg, int n) {
    int i = blockIdx.x * blockDim.x + threadIdx.x;
    if (i < n) deg[i] = 1.0f;            // self-loop weight 1
}

__global__ void k_accum_deg(float* deg, const int* dst, const float* ew, int e) {
    int i = blockIdx.x * blockDim.x + threadIdx.x;
    if (i < e) atomicAdd(&deg[dst[i]], ew[i]);
}

__global__ void k_dinv(float* deg, int n) {
    int i = blockIdx.x * blockDim.x + threadIdx.x;
    if (i < n) {
        float d = deg[i];
        deg[i] = (d > 0.0f) ? rsqrtf(d) : 0.0f;   // in place: deg -> dinv
    }
}

__global__ void k_norm(float* norm, const int* src, const int* dst,
                       const float* ew, const float* dinv, int e) {
    int i = blockIdx.x * blockDim.x + threadIdx.x;
    if (i < e) norm[i] = dinv[src[i]] * ew[i] * dinv[dst[i]];
}

// ---------------- WMMA fp32 GEMM: out[M,Nout] = A[M,K] @ W[K,Nout] ----------------
// One wave computes one 16x16 tile via V_WMMA_F32_16X16X4_F32 (K stepped by 4).
// 32-bit operand layouts (ISA 7.12.2): A/B per lane use K = k0 + 2*(lane>=16) + {0,1};
// C/D VGPR j -> row = 16*rowT + j + 8*(lane>=16), col = 16*colT + (lane&15).

__global__ void k_gemm_wmma(const float* __restrict__ A, const float* __restrict__ W,
                            float* __restrict__ out, const float* __restrict__ bias,
                            int M, int K, int Nout, int do_relu) {
    __shared__ float sW[F_IN * HID];           // 32 KB, covers all shapes used
    const int kn = K * Nout;
    for (int i = threadIdx.x; i < kn; i += blockDim.x) sW[i] = W[i];
    __syncthreads();

    const int nColT = Nout >> 4;
    const int totalTiles = (M >> 4) * nColT;
    const int tile = blockIdx.x * (blockDim.x >> 5) + (threadIdx.x >> 5);
    if (tile >= totalTiles) return;            // wave-uniform: EXEC stays all-ones below

    const int rowT = tile / nColT;
    const int colT = tile % nColT;
    const int lane = threadIdx.x & 31;
    const int half = lane >> 4;                // 0 or 1
    const int hi2  = half << 1;                // K offset for this lane half
    const int m    = (rowT << 4) + (lane & 15);
    const int n    = (colT << 4) + (lane & 15);

    const float* Arow = A + (size_t)m * K;
    v8f acc = {};
    for (int k0 = 0; k0 < K; k0 += 4) {
        const int kk = k0 + hi2;
        const float2 av = *(const float2*)(Arow + kk);     // 8B-aligned (kk even)
        v2f a; a.x = av.x; a.y = av.y;
        v2f b; b.x = sW[kk * Nout + n]; b.y = sW[(kk + 1) * Nout + n];
        acc = __builtin_amdgcn_wmma_f32_16x16x4_f32(
            /*neg_a=*/false, a, /*neg_b=*/false, b,
            /*c_mod=*/(short)0, acc, /*reuse_a=*/false, /*reuse_b=*/false);
    }

    const float bn = bias ? bias[n] : 0.0f;
    #pragma unroll
    for (int j = 0; j < 8; ++j) {
        const int r = (rowT << 4) + j + (half << 3);
        float v = acc[j] + bn;
        if (do_relu) v = v > 0.0f ? v : 0.0f;
        out[(size_t)r * Nout + n] = v;
    }
}

// ---------------- graph aggregation ----------------

// self-loop seed: aggB[i,:] = t[i,:] * dinv[i]^2   (float4 per thread, H=64 -> 16/node)
__global__ void k_seed_self(float* __restrict__ aggB, const float* __restrict__ t,
                            const float* __restrict__ dinv, int n) {
    int idx = blockIdx.x * blockDim.x + threadIdx.x;
    if (idx >= n * 16) return;
    int node = idx >> 4;
    float s = dinv[node]; s *= s;
    float4 v = ((const float4*)t)[idx];
    v.x *= s; v.y *= s; v.z *= s; v.w *= s;
    ((float4*)aggB)[idx] = v;
}

// edge scatter: aggB[dst,:] += t[src,:] * norm[e]   (float4 gather + 4 atomics)
__global__ void k_scatter(float* __restrict__ aggB, const float* __restrict__ t,
                          const int* __restrict__ src, const int* __restrict__ dst,
                          const float* __restrict__ norm, int e) {
    int idx = blockIdx.x * blockDim.x + threadIdx.x;
    if (idx >= e * 16) return;
    int ed = idx >> 4;
    int q  = idx & 15;
    int s = src[ed], d = dst[ed];
    float nm = norm[ed];
    float4 v = ((const float4*)(t + (size_t)s * HID))[q];
    float* p = aggB + (size_t)d * HID + (q << 2);
    atomicAdd(p + 0, v.x * nm);
    atomicAdd(p + 1, v.y * nm);
    atomicAdd(p + 2, v.z * nm);
    atomicAdd(p + 3, v.w * nm);
}

// in-place: h = relu(agg + b)
__global__ void k_bias_relu(float* __restrict__ h, const float* __restrict__ bias, int n) {
    int idx = blockIdx.x * blockDim.x + threadIdx.x;
    if (idx >= n * 16) return;
    int q = idx & 15;
    float4 v = ((float4*)h)[idx];
    const float4 b = ((const float4*)bias)[q];
    v.x = fmaxf(v.x + b.x, 0.0f);
    v.y = fmaxf(v.y + b.y, 0.0f);
    v.z = fmaxf(v.z + b.z, 0.0f);
    v.w = fmaxf(v.w + b.w, 0.0f);
    ((float4*)h)[idx] = v;
}

// final classifier layer: out[n,3] = hc[n,32] @ cW2[32,3] + cb2
__global__ void k_final(float* __restrict__ out, const float* __restrict__ hc,
                        const float* __restrict__ cW2, const float* __restrict__ cb2, int n) {
    int i = blockIdx.x * blockDim.x + threadIdx.x;
    if (i >= n) return;
    const float* row = hc + (size_t)i * 32;
    float acc0 = cb2[0], acc1 = cb2[1], acc2 = cb2[2];
    #pragma unroll
    for (int j = 0; j < 32; ++j) {
        float h = row[j];
        acc0 += h * cW2[j * 3 + 0];
        acc1 += h * cW2[j * 3 + 1];
        acc2 += h * cW2[j * 3 + 2];
    }
    out[(size_t)i * 3 + 0] = acc0;
    out[(size_t)i * 3 + 1] = acc1;
    out[(size_t)i * 3 + 2] = acc2;
}

// ---------------- host launcher ----------------

static inline unsigned cdiv(long long a, long long b) { return (unsigned)((a + b - 1) / b); }

extern "C" void kernel_launch(void* const* d_in, const int* in_sizes, int n_in,
                              void* d_out, int out_size, void* d_ws, size_t ws_size,
                              hipStream_t stream) {
    const float* x   = (const float*)d_in[0];
    const int*   ei  = (const int*)d_in[1];
    const float* ew  = (const float*)d_in[2];
    const float* W0  = (const float*)d_in[3];
    const float* b0  = (const float*)d_in[4];
    const float* W1  = (const float*)d_in[5];
    const float* b1  = (const float*)d_in[6];
    const float* cW1 = (const float*)d_in[7];
    const float* cb1 = (const float*)d_in[8];
    const float* cW2 = (const float*)d_in[9];
    const float* cb2 = (const float*)d_in[10];
    float* out = (float*)d_out;

    const int N = in_sizes[0] / F_IN;   // 50000 (multiple of 16)
    const int E = in_sizes[2];          // 800000
    const int* src = ei;
    const int* dst = ei + E;

    // workspace layout (floats): dinv[N] | norm[E] | bufA[N*64] | bufB[N*64]
    float* ws   = (float*)d_ws;
    float* dinv = ws;
    float* norm = ws + N;
    float* bufA = norm + E;
    float* bufB = bufA + (size_t)N * HID;

    const int B = 256;

    // ---- gcn_norm ----
    k_init_deg<<<cdiv(N, B), B, 0, stream>>>(dinv, N);
    k_accum_deg<<<cdiv(E, B), B, 0, stream>>>(dinv, dst, ew, E);
    k_dinv<<<cdiv(N, B), B, 0, stream>>>(dinv, N);
    k_norm<<<cdiv(E, B), B, 0, stream>>>(norm, src, dst, ew, dinv, E);

    // ---- conv0: bufA = x @ W0 ; aggregate into bufB ; relu(+b0) ----
    {
        unsigned tiles = (unsigned)(N / 16) * (HID / 16);
        k_gemm_wmma<<<cdiv(tiles, 8), B, 0, stream>>>(x, W0, bufA, nullptr, N, F_IN, HID, 0);
    }
    k_seed_self<<<cdiv((long long)N * 16, B), B, 0, stream>>>(bufB, bufA, dinv, N);
    k_scatter<<<cdiv((long long)E * 16, B), B, 0, stream>>>(bufB, bufA, src, dst, norm, E);
    k_bias_relu<<<cdiv((long long)N * 16, B), B, 0, stream>>>(bufB, b0, N);

    // ---- conv1: bufA = h0 @ W1 ; aggregate into bufB ; relu(+b1) ----
    {
        unsigned tiles = (unsigned)(N / 16) * (HID / 16);
        k_gemm_wmma<<<cdiv(tiles, 8), B, 0, stream>>>(bufB, W1, bufA, nullptr, N, HID, HID, 0);
    }
    k_seed_self<<<cdiv((long long)N * 16, B), B, 0, stream>>>(bufB, bufA, dinv, N);
    k_scatter<<<cdiv((long long)E * 16, B), B, 0, stream>>>(bufB, bufA, src, dst, norm, E);
    k_bias_relu<<<cdiv((long long)N * 16, B), B, 0, stream>>>(bufB, b1, N);

    // ---- classifier: bufA[N,32] = relu(h1 @ cW1 + cb1) ; out = bufA @ cW2 + cb2 ----
    {
        unsigned tiles = (unsigned)(N / 16) * (32 / 16);
        k_gemm_wmma<<<cdiv(tiles, 8), B, 0, stream>>>(bufB, cW1, bufA, cb1, N, HID, 32, 1);
    }
    k_final<<<cdiv(N, B), B, 0, stream>>>(out, bufA, cW2, cb2, N);
}